// TripletLoss_26680336843539
// MI455X (gfx1250) — compile-verified
//
#include <hip/hip_runtime.h>
#include <math.h>
#include <stdint.h>

#define N_PTS 8192
#define DIM   128
#define MARGIN 0.3f
#define CSPLIT 4
#define COLS_PER_BLOCK (N_PTS / CSPLIT)        // 2048
#define PANEL_COLS 32
#define NPANEL (COLS_PER_BLOCK / PANEL_COLS)   // 64
#define LDS_STRIDE 132                         // 128 + 4 pad: bank-conflict-free

typedef float v2f __attribute__((ext_vector_type(2)));
typedef float v8f __attribute__((ext_vector_type(8)));

#if defined(__HIP_DEVICE_COMPILE__)
#if !__has_builtin(__builtin_amdgcn_wmma_f32_16x16x4_f32)
#error "missing __builtin_amdgcn_wmma_f32_16x16x4_f32 for gfx1250 (device pass)"
#endif
#endif

#if defined(__HIP_DEVICE_COMPILE__) && __has_builtin(__builtin_amdgcn_s_wait_asynccnt)
#define WAIT_ASYNC() __builtin_amdgcn_s_wait_asynccnt(0)
#else
#define WAIT_ASYNC() asm volatile("s_wait_asynccnt 0x0" ::: "memory")
#endif

// Kernel 1: squared row norms (one wave32 per row) + init mining partials + out.
__global__ void __launch_bounds__(256)
init_kernel(const float* __restrict__ X, float* __restrict__ sq,
            int* __restrict__ apInt, int* __restrict__ anInt,
            float* __restrict__ out) {
  if (blockIdx.x == 0 && threadIdx.x == 0) out[0] = 0.0f;
  const int gid  = blockIdx.x * blockDim.x + threadIdx.x;
  const int row  = gid >> 5;
  const int lane = gid & 31;
  if (row < N_PTS) {
    const float4 v = *(const float4*)(X + (size_t)row * DIM + lane * 4);
    float s = v.x * v.x + v.y * v.y + v.z * v.z + v.w * v.w;
#pragma unroll
    for (int off = 16; off > 0; off >>= 1) s += __shfl_xor(s, off, 32);
    if (lane == 0) {
      sq[row]    = s;
      apInt[row] = 0;           // max identity for non-negative floats
      anInt[row] = 0x7F7FFFFF;  // FLT_MAX bits: min identity
    }
  }
}

// Async DMA of a 32-col x 128-k fp32 panel: 1024 x 16B, 8 per thread, no VGPR
// staging (global_load_async_to_lds_b128 writes LDS directly, ASYNCcnt-tracked).
__device__ __forceinline__ void
async_panel_load(const float* __restrict__ X, int colstart,
                 float* __restrict__ ldsbase, int tid) {
#pragma unroll
  for (int s = 0; s < 8; ++s) {
    const int slot = tid + s * 128;
    const int col  = slot >> 5;
    const int k4   = (slot & 31) << 2;
    const float* g = X + (size_t)(colstart + col) * DIM + k4;
    // LDS aperture lives in addr[63:32]; low 32 bits are the LDS byte address.
    const unsigned l = (unsigned)(uintptr_t)(ldsbase + col * LDS_STRIDE + k4);
    asm volatile("global_load_async_to_lds_b128 %0, %1, off"
                 :: "v"(l), "v"(g) : "memory");
  }
}

// Kernel 2: fused Gram (f32 WMMA) + batch-hard mining over a column chunk.
// Block = 128 thr = 4 waves; wave owns 16 rows, 2 column tiles per 32-col panel.
// Double-buffered LDS fed by async global->LDS DMA; one barrier per panel.
__global__ void __launch_bounds__(128, 1)
gram_mine_kernel(const float* __restrict__ X, const int* __restrict__ tgt,
                 const float* __restrict__ sq, int* __restrict__ apInt,
                 int* __restrict__ anInt) {
  __shared__ float lds[2][PANEL_COLS * LDS_STRIDE];  // 2 x 16.9 KB

  const int tid  = threadIdx.x;
  const int wave = tid >> 5;
  const int lane = tid & 31;
  const int mrow = lane & 15;   // A row-in-tile; also B/C/D column-in-tile
  const int kgrp = lane >> 4;

  const int rowbase  = blockIdx.x * 64 + wave * 16;
  const int colchunk = blockIdx.y * COLS_PER_BLOCK;

  // A fragment: 16x128 row panel resident in 64 VGPRs for the whole kernel.
  v2f aFrag[32];
  {
    const float* aptr = X + (size_t)(rowbase + mrow) * DIM + kgrp * 2;
#pragma unroll
    for (int t = 0; t < 32; ++t) aFrag[t] = *(const v2f*)(aptr + t * 4);
  }

  // C/D layout: lane, VGPR v -> (row = v + 8*kgrp, col = mrow [+16 for tile1]).
  int   trow[8];
  float sqrow[8];
#pragma unroll
  for (int v = 0; v < 8; ++v) {
    const int r = rowbase + v + 8 * kgrp;
    trow[v]  = tgt[r];
    sqrow[v] = sq[r];
  }

  float apos[8], aneg[8];
#pragma unroll
  for (int v = 0; v < 8; ++v) { apos[v] = 0.0f; aneg[v] = 3.4e38f; }

  // Prologue: DMA panel 0 into buffer 0.
  async_panel_load(X, colchunk, &lds[0][0], tid);
  WAIT_ASYNC();
  __syncthreads();

  for (int p = 0; p < NPANEL; ++p) {
    const int buf     = p & 1;
    const int colbase = colchunk + p * PANEL_COLS;
    const bool more   = (p + 1) < NPANEL;

    // Kick off DMA of the next panel into the idle buffer (safe: the barrier
    // ending iteration p-1 guaranteed all waves finished reading buf^1).
    if (more) async_panel_load(X, colbase + PANEL_COLS, &lds[buf ^ 1][0], tid);

    // Two 16x16 Gram tiles, K=128: 32 DS-read bursts feeding 64 WMMAs.
    v8f c0 = {}, c1 = {};
    const float* bp0 = &lds[buf][mrow * LDS_STRIDE + kgrp * 2];
    const float* bp1 = bp0 + 16 * LDS_STRIDE;
#pragma unroll
    for (int ch = 0; ch < 4; ++ch) {
      v2f b0[8], b1[8];
#pragma unroll
      for (int t = 0; t < 8; ++t) {
        b0[t] = *(const v2f*)(bp0 + (ch * 8 + t) * 4);
        b1[t] = *(const v2f*)(bp1 + (ch * 8 + t) * 4);
      }
#pragma unroll
      for (int t = 0; t < 8; ++t) {
        c0 = __builtin_amdgcn_wmma_f32_16x16x4_f32(false, aFrag[ch * 8 + t],
                                                   false, b0[t], (short)0, c0,
                                                   false, false);
        c1 = __builtin_amdgcn_wmma_f32_16x16x4_f32(false, aFrag[ch * 8 + t],
                                                   false, b1[t], (short)0, c1,
                                                   false, false);
      }
    }

#if defined(__HIP_DEVICE_COMPILE__) && __has_builtin(__builtin_amdgcn_sched_group_barrier)
    // Software pipeline the region: 8 x [4 DS reads || 8 WMMA] so each LDS
    // round-trip is covered by the previous group's 8 matrix ops instead of 2.
#pragma unroll
    for (int g = 0; g < 8; ++g) {
      __builtin_amdgcn_sched_group_barrier(0x100, 4, 0);  // DS read group
      __builtin_amdgcn_sched_group_barrier(0x008, 8, 0);  // WMMA group
    }
#endif

    const int   tc0 = tgt[colbase + mrow];
    const int   tc1 = tgt[colbase + 16 + mrow];
    const float sc0 = sq[colbase + mrow];
    const float sc1 = sq[colbase + 16 + mrow];
#pragma unroll
    for (int v = 0; v < 8; ++v) {
      float d0 = fmaxf(fmaf(-2.0f, c0[v], sqrow[v] + sc0), 0.0f);
      float d1 = fmaxf(fmaf(-2.0f, c1[v], sqrow[v] + sc1), 0.0f);
      const bool s0 = (trow[v] == tc0);
      const bool s1 = (trow[v] == tc1);
      apos[v] = (s0  && d0 > apos[v]) ? d0 : apos[v];
      aneg[v] = (!s0 && d0 < aneg[v]) ? d0 : aneg[v];
      apos[v] = (s1  && d1 > apos[v]) ? d1 : apos[v];
      aneg[v] = (!s1 && d1 < aneg[v]) ? d1 : aneg[v];
    }

    // My DMA writes have landed in LDS; barrier publishes everyone's panel and
    // retires all reads of lds[buf] before it is overwritten next iteration.
    WAIT_ASYNC();
    __syncthreads();
  }

  // Row reduction across the 16 lanes of each half-wave.
#pragma unroll
  for (int v = 0; v < 8; ++v) {
#pragma unroll
    for (int off = 8; off > 0; off >>= 1) {
      apos[v] = fmaxf(apos[v], __shfl_xor(apos[v], off, 32));
      aneg[v] = fminf(aneg[v], __shfl_xor(aneg[v], off, 32));
    }
  }

  // Merge partials across column chunks: non-negative floats order as ints.
  if (mrow == 0) {
#pragma unroll
    for (int v = 0; v < 8; ++v) {
      const int r = rowbase + v + 8 * kgrp;
      atomicMax(&apInt[r], __float_as_int(apos[v]));
      atomicMin(&anInt[r], __float_as_int(aneg[v]));
    }
  }
}

// Kernel 3: per-row margin loss + mean.
__global__ void __launch_bounds__(256)
loss_kernel(const int* __restrict__ apInt, const int* __restrict__ anInt,
            float* __restrict__ out) {
  const int row = blockIdx.x * blockDim.x + threadIdx.x;
  const float ap = __int_as_float(apInt[row]);
  const float an = __int_as_float(anInt[row]);
  float l = fmaxf(sqrtf(ap) - sqrtf(an) + MARGIN, 0.0f);
#pragma unroll
  for (int off = 16; off > 0; off >>= 1) l += __shfl_xor(l, off, 32);
  if ((threadIdx.x & 31) == 0) atomicAdd(out, l * (1.0f / (float)N_PTS));
}

extern "C" void kernel_launch(void* const* d_in, const int* in_sizes, int n_in,
                              void* d_out, int out_size, void* d_ws, size_t ws_size,
                              hipStream_t stream) {
  const float* X   = (const float*)d_in[0];
  const int*   tgt = (const int*)d_in[1];
  float* out = (float*)d_out;

  float* sq    = (float*)d_ws;           // 8192 floats
  int*   apInt = (int*)(sq + N_PTS);     // 8192 ints
  int*   anInt = (int*)(apInt + N_PTS);  // 8192 ints  (total 96 KB scratch)

  init_kernel<<<(N_PTS * 32) / 256, 256, 0, stream>>>(X, sq, apInt, anInt, out);
  gram_mine_kernel<<<dim3(N_PTS / 64, CSPLIT), 128, 0, stream>>>(X, tgt, sq,
                                                                 apInt, anInt);
  loss_kernel<<<N_PTS / 256, 256, 0, stream>>>(apInt, anInt, out);
}